// SSTDIS_12833362280505
// MI455X (gfx1250) — compile-verified
//
#include <hip/hip_runtime.h>
#include <hip/hip_bf16.h>

// Problem constants from the reference: B=64, E=64, D=128
#define NROWS 8192      // 2*B*E flattened rows
#define DDIM  128
#define HALF_ROWS 4096  // rows from feature_k; rest from feature_s
#define LOG2E 1.4426950408889634f
#define LDS_ROW 132     // padded LDS row stride (floats): 16B-aligned, bank-conflict-free

typedef __attribute__((ext_vector_type(2))) float v2f;
typedef __attribute__((ext_vector_type(8))) float v8f;

__device__ __forceinline__ const float* row_ptr(const float* fk, const float* fs, int r) {
    return (r < HALF_ROWS) ? (fk + (size_t)r * DDIM)
                           : (fs + (size_t)(r - HALF_ROWS) * DDIM);
}

// Sum across each 16-lane half of a wave32 (xor masks 1,2,4,8 stay within a half).
__device__ __forceinline__ float half16_reduce(float v) {
    v += __shfl_xor(v, 1, 32);
    v += __shfl_xor(v, 2, 32);
    v += __shfl_xor(v, 4, 32);
    v += __shfl_xor(v, 8, 32);
    return v;
}

// ---------------------------------------------------------------------------
// Kernel 1: inverse L2 norms. One wave per row (4 floats per lane).
// ---------------------------------------------------------------------------
__global__ __launch_bounds__(128) void rownorm_kernel(
    const float* __restrict__ fk, const float* __restrict__ fs,
    float* __restrict__ invn)
{
    const int lane = threadIdx.x & 31;
    const int wave = threadIdx.x >> 5;
    const int r = blockIdx.x * 4 + wave;           // grid.x = 2048
    const float* row = row_ptr(fk, fs, r);
    const float4 v = *(const float4*)(row + lane * 4);
    float ss = v.x * v.x + v.y * v.y + v.z * v.z + v.w * v.w;
    ss += __shfl_xor(ss, 1, 32);
    ss += __shfl_xor(ss, 2, 32);
    ss += __shfl_xor(ss, 4, 32);
    ss += __shfl_xor(ss, 8, 32);
    ss += __shfl_xor(ss, 16, 32);
    if (lane == 0) invn[r] = 1.0f / fmaxf(sqrtf(ss), 1e-12f);
}

// ---------------------------------------------------------------------------
// Kernel 2: fused C = exp(F F^T) with row reductions. WMMA on RAW data, scales
// applied in the epilogue. B tiles are staged to per-wave LDS double buffers
// with global_load_async_to_lds_b128 (ASYNCcnt-tracked), so a full tile of
// WMMA work hides the global latency; fragments come from LDS (ds_load_b64).
// grid = (8 col-supertiles, 512 row-tiles), block = 128 threads (4 waves).
// Wave w handles col tiles ct = 64*sx + 4*it + w, it = 0..15.
// ---------------------------------------------------------------------------
__global__ __launch_bounds__(128) void gemm_exp_reduce_kernel(
    const float* __restrict__ fk, const float* __restrict__ fs,
    const float* __restrict__ invn,
    float* __restrict__ pTot, float* __restrict__ pPos)
{
    __shared__ float ldsB[4][2][16 * LDS_ROW];   // per-wave double buffer (~66 KB)
    __shared__ float sT[4][16];
    __shared__ float sP[4][16];

    const int lane = threadIdx.x & 31;
    const int wave = threadIdx.x >> 5;   // 0..3
    const int l15  = lane & 15;
    const int hi   = lane >> 4;          // 0: K+{0,1}, 1: K+{2,3} (ISA 16x4 fp32 layout)
    const int ry   = blockIdx.y;         // row tile 0..511
    const int sx   = blockIdx.x;         // col supertile 0..7

    const int rowBase = ry * 16;
    const int rm = rowBase + l15;                  // this lane's A row (M = l15)
    const float invA = invn[rm];
    const float* aRow = row_ptr(fk, fs, rm) + 2 * hi;

    // Preload RAW A stripe (16 rows x 128 K) as 32 K-step fragments (64 VGPRs).
    v2f A[32];
#pragma unroll
    for (int s = 0; s < 32; ++s)
        A[s] = *(const v2f*)(aRow + 4 * s);

    // Row scales for the C layout: lane's VGPR v holds row m = v + 8*hi.
    float invAm[8];
#pragma unroll
    for (int v = 0; v < 8; ++v)
        invAm[v] = __shfl(invA, v + 8 * hi, 32) * LOG2E;

    const int iblk    = ry >> 2;                   // i = r / 64
    const int pairBlk = (iblk + 64) & 127;         // (i + B) % 2B
    const int posTile = pairBlk * 4 + (ry & 3);    // col tile holding pos2 diagonal
    const int diagv   = l15 - 8 * hi;              // C diag (m==n) at lanes v and 24+v

    // Async staging addresses: lane covers row n = l15, float4-column pair hi.
    // Same INST_OFFSET (i*32 B) advances both global and LDS sides (ISA 10.x).
    const unsigned gDelta = (unsigned)(l15 * (DDIM * 4) + hi * 16);
    const unsigned lDelta = (unsigned)(l15 * (LDS_ROW * 4) + hi * 16);
    const unsigned ldsBase[2] = {
        (unsigned)(uintptr_t)(void*)&ldsB[wave][0][0] + lDelta,
        (unsigned)(uintptr_t)(void*)&ldsB[wave][1][0] + lDelta
    };

    auto issue_tile = [&](int it, int buf) {
        const int ct = sx * 64 + it * 4 + wave;
        const float* tb = row_ptr(fk, fs, ct * 16);         // tile is contiguous 8 KB
        const unsigned long long ga = (unsigned long long)(uintptr_t)tb + gDelta;
        const unsigned la = ldsBase[buf];
#pragma unroll
        for (int i = 0; i < 16; ++i) {
            asm volatile("global_load_async_to_lds_b128 %0, %1, off offset:%2"
                         :: "v"(la), "v"(ga), "n"(i * 32) : "memory");
        }
    };

    v8f tot = {};
    v8f pos = {};

    issue_tile(0, 0);                              // prologue: tile 0 in flight

    for (int it = 0; it < 16; ++it) {
        if (it < 15) {
            issue_tile(it + 1, (it + 1) & 1);      // prefetch next tile
            // In-order async completion: <=16 outstanding => tile `it` resident.
            asm volatile("s_wait_asynccnt 16" ::: "memory");
        } else {
            asm volatile("s_wait_asynccnt 0" ::: "memory");
        }

        const int ct = sx * 64 + it * 4 + wave;
        const int cn = ct * 16 + l15;              // this lane's B column (N = l15)
        const float invB = invn[cn];
        const int bufRow = (it & 1) * 0 + l15 * LDS_ROW + 2 * hi;  // base (floats)
        const float* lbuf = &ldsB[wave][it & 1][0];

        v8f c = {};
#pragma unroll
        for (int chunk = 0; chunk < 4; ++chunk) {
            v2f Bf[8];
#pragma unroll
            for (int q = 0; q < 8; ++q) {
                const int s = chunk * 8 + q;
                Bf[q] = *(const v2f*)(lbuf + bufRow + 4 * s);
            }
#pragma unroll
            for (int q = 0; q < 8; ++q)
                c = __builtin_amdgcn_wmma_f32_16x16x4_f32(
                        false, A[chunk * 8 + q], false, Bf[q],
                        (short)0, c, false, false);
        }

        const bool isPos1 = ((ct >> 2) == iblk);   // same-i block: whole tile counts
        const bool isPos2 = (ct == posTile);       // cross-view: diagonal only
#pragma unroll
        for (int v = 0; v < 8; ++v) {
            // exp(dot*invA*invB) = exp2(dot * (invA*log2e) * invB)
            const float e = exp2f(c[v] * invAm[v] * invB);
            tot[v] += e;
            if (isPos1) pos[v] += e;
            if (isPos2 && diagv == v) pos[v] += e;
        }
    }

    // Per-row sums: reduce across the 16 lanes of each half (rows v+8*hi).
#pragma unroll
    for (int v = 0; v < 8; ++v) {
        tot[v] = half16_reduce(tot[v]);
        pos[v] = half16_reduce(pos[v]);
    }

    if (l15 == 0) {
#pragma unroll
        for (int v = 0; v < 8; ++v) {
            sT[wave][v + 8 * hi] = tot[v];
            sP[wave][v + 8 * hi] = pos[v];
        }
    }
    __syncthreads();

    // Combine the block's 4 waves (fixed order -> deterministic), one writer/slot.
    if (threadIdx.x < 16) {
        const int m = threadIdx.x;
        const float t = sT[0][m] + sT[1][m] + sT[2][m] + sT[3][m];
        const float p = sP[0][m] + sP[1][m] + sP[2][m] + sP[3][m];
        const int r = rowBase + m;
        pTot[r * 8 + sx] = t;
        pPos[r * 8 + sx] = p;
    }
}

// ---------------------------------------------------------------------------
// Kernel 3: loss = mean(log(total) - log(pos)). Single block, fixed-order sums.
// ---------------------------------------------------------------------------
__global__ __launch_bounds__(256) void loss_kernel(
    const float* __restrict__ pTot, const float* __restrict__ pPos,
    float* __restrict__ out)
{
    float acc = 0.0f;
    for (int r = threadIdx.x; r < NROWS; r += 256) {
        float t = 0.0f, p = 0.0f;
#pragma unroll
        for (int x = 0; x < 8; ++x) {
            t += pTot[r * 8 + x];
            p += pPos[r * 8 + x];
        }
        acc += logf(t) - logf(p);
    }
    __shared__ float s[256];
    s[threadIdx.x] = acc;
    __syncthreads();
    for (int off = 128; off > 0; off >>= 1) {
        if (threadIdx.x < off) s[threadIdx.x] += s[threadIdx.x + off];
        __syncthreads();
    }
    if (threadIdx.x == 0) out[0] = s[0] / (float)NROWS;
}

// ---------------------------------------------------------------------------
extern "C" void kernel_launch(void* const* d_in, const int* in_sizes, int n_in,
                              void* d_out, int out_size, void* d_ws, size_t ws_size,
                              hipStream_t stream) {
    const float* fk = (const float*)d_in[0];   // [64, 64, 128] fp32
    const float* fs = (const float*)d_in[1];   // [64, 64, 128] fp32

    float* ws   = (float*)d_ws;
    float* invn = ws;                          // 8192 floats
    float* pTot = ws + NROWS;                  // 8192*8 floats
    float* pPos = ws + NROWS + NROWS * 8;      // 8192*8 floats  (~548 KB total)

    rownorm_kernel<<<dim3(NROWS / 4), dim3(128), 0, stream>>>(fk, fs, invn);

    dim3 grid(8, NROWS / 16);                  // 8 col supertiles x 512 row tiles
    gemm_exp_reduce_kernel<<<grid, dim3(128), 0, stream>>>(fk, fs, invn, pTot, pPos);

    loss_kernel<<<dim3(1), dim3(256), 0, stream>>>(pTot, pPos, (float*)d_out);
}